// MPSPeriodic_16527034155479
// MI455X (gfx1250) — compile-verified
//
#include <hip/hip_runtime.h>
#include <hip/hip_bf16.h>
#include <math.h>

// Problem constants (match reference): B=256, L=128, d=2, D=128
#define SITES   128
#define LOCALD  2
#define BOND    128
#define ESTRIDE 132                 // edge row stride (floats); 132 % 64 == 4
#define NPITCH  68                  // site-matrix per-column pitch;  68 % 64 == 4
#define HOFF    (BOND * NPITCH)     // 8704 floats between k-half planes (== 0 mod 64)
#define E_FLOATS (BOND * ESTRIDE)   // 16896
#define A_FLOATS (2 * HOFF)         // 17408

typedef float v2f __attribute__((ext_vector_type(2)));
typedef float v4f __attribute__((ext_vector_type(4)));
typedef float v8f __attribute__((ext_vector_type(8)));

// permuted column index inside an edge row: frag-quad native order
__device__ __forceinline__ int pcol(int k) {
    return ((k >> 1) & 1) * 64 + (k >> 2) * 2 + (k & 1);
}

extern __shared__ float smem[];   // [E_FLOATS] edge | [A_FLOATS] interleaved site matrix

__global__ __launch_bounds__(256) void mps_chain_wmma_f32(
    const int*   __restrict__ x,        // (B, L) int32 indices in [0, d)
    const float* __restrict__ tensors,  // (L, d, D, D) fp32
    float*       __restrict__ out)      // (B,) complex64 -> 2*B floats
{
    float* Elds = smem;
    float* Abt  = smem + E_FLOATS;  // A[k][n] at h*HOFF + n*NPITCH + 2*(k>>2) + (k&1)

    const int b    = blockIdx.x;
    const int tid  = threadIdx.x;
    const int wave = tid >> 5;          // 0..7, owns edge rows [16*wave, 16*wave+16)
    const int lane = tid & 31;
    const int half = lane >> 4;         // K-half selector in A/B frags
    const int lm   = lane & 15;         // m for A-frag, n for B-frag

    // ---- init edge = identity (in permuted-column layout) ----
    #pragma unroll
    for (int r = 0; r < 66; ++r) {
        int idx = r * 256 + tid;
        if (idx < E_FLOATS) Elds[idx] = 0.0f;
    }
    __syncthreads();
    if (tid < BOND) Elds[tid * ESTRIDE + pcol(tid)] = 1.0f;

    // A-frag base: E[m=16w+lm], k-half plane `half`; b128 at +4*ktp -> kt=2ktp,2ktp+1
    const float* aBase = &Elds[(16 * wave + lm) * ESTRIDE + half * 64];
    // B-frag base: column n=lm (+16*nt), k-half plane `half`
    const float* bBase = &Abt[half * HOFF + lm * NPITCH];
    // writeback: per-lane constant permuted column offset for n = 16*nt + lm
    const int pc0   = ((lm >> 1) & 1) * 64 + 2 * (lm >> 2) + (lm & 1);
    float* wbBase   = &Elds[(16 * wave + 8 * half) * ESTRIDE + pc0];

    for (int l = 0; l < SITES; ++l) {
        // ---- stage A = tensors[l, x[b,l]] into frag-native LDS layout ----
        __syncthreads();   // prior step done reading Abt
        const int q = x[(size_t)b * SITES + l];
        const float4* g = reinterpret_cast<const float4*>(
            tensors + ((size_t)l * LOCALD + q) * (BOND * BOND));
        #pragma unroll
        for (int r = 0; r < 16; ++r) {
            int f4 = r * 256 + tid;          // 0..4095
            float4 v = g[f4];
            int k = f4 >> 5;                 // row
            int n = (f4 & 31) * 4;           // first of 4 columns
            int kb = ((k >> 1) & 1) * HOFF + 2 * (k >> 2) + (k & 1);
            Abt[kb + (n + 0) * NPITCH] = v.x;
            Abt[kb + (n + 1) * NPITCH] = v.y;
            Abt[kb + (n + 2) * NPITCH] = v.z;
            Abt[kb + (n + 3) * NPITCH] = v.w;
        }
        __syncthreads();

        // ---- C(rows of this wave) = E(rows of this wave) @ A ----
        v8f acc[8];
        #pragma unroll
        for (int nt = 0; nt < 8; ++nt) acc[nt] = (v8f){0,0,0,0,0,0,0,0};

        // fully unrolled, pipelined over kt-pairs: one b128 covers two K-tiles
        v4f aCur = *reinterpret_cast<const v4f*>(aBase);
        v4f bCur[8];
        #pragma unroll
        for (int nt = 0; nt < 8; ++nt)
            bCur[nt] = *reinterpret_cast<const v4f*>(bBase + 1088 * nt);

        #pragma unroll
        for (int ktp = 0; ktp < 16; ++ktp) {
            v4f aNxt = aCur;
            v4f bNxt[8];
            if (ktp < 15) {
                aNxt = *reinterpret_cast<const v4f*>(aBase + 4 * (ktp + 1));
                #pragma unroll
                for (int nt = 0; nt < 8; ++nt)
                    bNxt[nt] = *reinterpret_cast<const v4f*>(
                        bBase + 1088 * nt + 4 * (ktp + 1));
            }
            v2f aLo = __builtin_shufflevector(aCur, aCur, 0, 1);
            v2f aHi = __builtin_shufflevector(aCur, aCur, 2, 3);
            #pragma unroll
            for (int nt = 0; nt < 8; ++nt) {
                v2f bLo = __builtin_shufflevector(bCur[nt], bCur[nt], 0, 1);
                acc[nt] = __builtin_amdgcn_wmma_f32_16x16x4_f32(
                    false, aLo, false, bLo, (short)0, acc[nt], false, false);
            }
            #pragma unroll
            for (int nt = 0; nt < 8; ++nt) {
                v2f bHi = __builtin_shufflevector(bCur[nt], bCur[nt], 2, 3);
                acc[nt] = __builtin_amdgcn_wmma_f32_16x16x4_f32(
                    false, aHi, false, bHi, (short)0, acc[nt], false, false);
            }
            if (ktp < 15) {
                aCur = aNxt;
                #pragma unroll
                for (int nt = 0; nt < 8; ++nt) bCur[nt] = bNxt[nt];
            }
        }

        // ---- write slice back in place (per-wave LDS ordering makes it safe) ----
        #pragma unroll
        for (int nt = 0; nt < 8; ++nt) {
            #pragma unroll
            for (int vr = 0; vr < 8; ++vr) {
                // D layout: m = 16w + vr + 8*half, permuted col = pc0 + 8*nt
                wbBase[vr * ESTRIDE + 8 * nt] = acc[nt][vr];
            }
        }
    }

    // ---- psi = trace(E); out = log(psi) as complex64 ----
    __syncthreads();
    float* red = Abt;                    // reuse staging buffer as scratch
    if (tid < BOND) red[tid] = Elds[tid * ESTRIDE + pcol(tid)];
    __syncthreads();
    if (tid == 0) {
        float psi = 0.0f;
        #pragma unroll 8
        for (int i = 0; i < BOND; ++i) psi += red[i];
        out[2 * b + 0] = logf(fabsf(psi));
        out[2 * b + 1] = (psi < 0.0f) ? 3.14159265358979323846f : 0.0f;
    }
}

extern "C" void kernel_launch(void* const* d_in, const int* in_sizes, int n_in,
                              void* d_out, int out_size, void* d_ws, size_t ws_size,
                              hipStream_t stream) {
    const int*   x       = (const int*)d_in[0];    // (B, L) int32
    const float* tensors = (const float*)d_in[1];  // (L, d, D, D) fp32
    float*       out     = (float*)d_out;          // B complex64 values

    const int B = in_sizes[0] / SITES;
    const size_t shmem = (size_t)(E_FLOATS + A_FLOATS) * sizeof(float); // ~134 KB

    mps_chain_wmma_f32<<<B, 256, shmem, stream>>>(x, tensors, out);
}